// Bottleneck_34479997452293
// MI455X (gfx1250) — compile-verified
//
#include <hip/hip_runtime.h>
#include <hip/hip_bf16.h>

// ---------------------------------------------------------------------------
// Problem constants (from reference)
// ---------------------------------------------------------------------------
static constexpr int B = 8, C = 256, E = 512, T = 8192, K = 512;
static constexpr int MAXCOLS = 512;            // at most K distinct t-columns per batch
static constexpr int BLK_PER_BATCH = (T / 64) * (E / 32);  // GEMM blocks per batch = 2048

typedef float v2f __attribute__((ext_vector_type(2)));
typedef float v8f __attribute__((ext_vector_type(8)));

// Order-preserving float<->uint encoding (ascending)
__device__ __forceinline__ unsigned ordenc(float f) {
  unsigned u = __float_as_uint(f);
  return (u & 0x80000000u) ? ~u : (u | 0x80000000u);
}
__device__ __forceinline__ float orddec(unsigned e) {
  unsigned u = (e & 0x80000000u) ? (e & 0x7FFFFFFFu) : ~e;
  return __uint_as_float(u);
}

// CDNA5 async global->LDS copy (ASYNCcnt path). LDS dest address is the low
// 32 bits of the flat pointer.
__device__ __forceinline__ void async_load_b128(unsigned lds_off, const void* gaddr) {
  asm volatile("global_load_async_to_lds_b128 %0, %1, off"
               :: "v"(lds_off), "v"(gaddr)
               : "memory");
}
__device__ __forceinline__ void wait_asynccnt0() {
  asm volatile("s_wait_asynccnt 0x0" ::: "memory");
}

// ---------------------------------------------------------------------------
// Generic fill (zero/seed scratch deterministically every launch)
// ---------------------------------------------------------------------------
__global__ void fill_u32(unsigned* p, size_t n, unsigned v) {
  for (size_t i = (size_t)blockIdx.x * blockDim.x + threadIdx.x; i < n;
       i += (size_t)gridDim.x * blockDim.x)
    p[i] = v;
}

// ---------------------------------------------------------------------------
// 1) invn[b,t] = 1 / (||x[b,:,t]|| + 1e-8)
// ---------------------------------------------------------------------------
__global__ void norms_kernel(const float* __restrict__ x, float* __restrict__ invn) {
  int i = blockIdx.x * 256 + threadIdx.x;  // over B*T (t fastest -> coalesced)
  if (i >= B * T) return;
  int b = i / T, t = i % T;
  const float* xp = x + (size_t)b * C * T + t;
  float s = 0.f;
  for (int c = 0; c < C; ++c) { float v = xp[(size_t)c * T]; s += v * v; }
  invn[i] = 1.0f / (sqrtf(s) + 1e-8f);
}

// ---------------------------------------------------------------------------
// 2) m[b,e,t] = invn[b,t] * sum_c mask_w[e,c]*x[b,c,t] + mask_b[e]
//    (invn factors out of the K-sum, so x is staged RAW via async->LDS).
//    WMMA f32 16x16x4, WG = 8 waves, tile 32(E) x 64(T), K chunked by 64.
//    Fused epilogue: per-batch max (encoded atomicMax), per-block
//    (blockmax, sum exp(v-blockmax)) records, and pass-0 radix histogram.
// ---------------------------------------------------------------------------
__global__ __launch_bounds__(256)
void mask_gemm_kernel(const float* __restrict__ x, const float* __restrict__ mask_w,
                      const float* __restrict__ mask_b, const float* __restrict__ invn,
                      float* __restrict__ m, unsigned* __restrict__ bmax_enc,
                      unsigned* __restrict__ hist, float* __restrict__ blkmax,
                      float* __restrict__ blksum) {
  const int b  = blockIdx.z;
  const int e0 = blockIdx.y * 32;
  const int t0 = blockIdx.x * 64;

  __shared__ __align__(16) float xs[64][64];  // [k][t] raw x chunk (16 KB)
  __shared__ float wsm[32][68];   // [e][k] mask_w chunk, pad 68 -> conflict-free A reads
  __shared__ float invt[64];
  __shared__ float mbias[32];
  __shared__ float red[256];
  __shared__ unsigned h[256];

  const int tid  = threadIdx.x;
  const int wave = tid >> 5, lane = tid & 31;
  const int er = (wave >> 2) * 16;  // 0 | 16  (E offset of this wave's 16x16)
  const int tc = (wave & 3) * 16;   // 0,16,32,48 (T offset)

  if (tid < 64) invt[tid] = invn[b * T + t0 + tid];
  if (tid < 32) mbias[tid] = mask_b[e0 + tid];
  h[tid] = 0;

  v8f acc = {};
  const int mrow  = lane & 15;
  const int khalf = (lane >> 4) * 2;  // lanes 0-15 -> K 0/1, lanes 16-31 -> K 2/3

  for (int c0 = 0; c0 < C; c0 += 64) {
    __syncthreads();
    // x tile: 64x64 floats = 1024 x B128 chunks, 4 per thread, async -> LDS
    {
      const float* gbase = x + ((size_t)b * C + c0) * T + t0;
#pragma unroll
      for (int it = 0; it < 4; ++it) {
        int ci  = it * 256 + tid;   // 0..1023
        int kk  = ci >> 4;          // row 0..63
        int tch = (ci & 15) * 4;    // col 0,4,...,60
        async_load_b128((unsigned)(uintptr_t)&xs[kk][tch],
                        gbase + (size_t)kk * T + tch);
      }
    }
    // weight tile: 32x64, regular stores
    for (int i = tid; i < 32 * 64; i += 256) {
      int e2 = i >> 6, kk = i & 63;
      wsm[e2][kk] = mask_w[(size_t)(e0 + e2) * C + c0 + kk];
    }
    wait_asynccnt0();    // drain this wave's async LDS writes (barrier only covers DS)
    __syncthreads();
    for (int k = 0; k < 64; k += 4) {
      v2f a, bb;
      a.x  = wsm[er + mrow][k + khalf];
      a.y  = wsm[er + mrow][k + khalf + 1];
      bb.x = xs[k + khalf][tc + mrow];
      bb.y = xs[k + khalf + 1][tc + mrow];
      acc = __builtin_amdgcn_wmma_f32_16x16x4_f32(false, a, false, bb,
                                                  (short)0, acc, false, false);
    }
  }

  // ---- epilogue ----
  // D layout: VGPR r -> M = r + (lane>=16 ? 8 : 0), N = lane&15
  const int ncol  = lane & 15;
  const int mhalf = (lane >> 4) * 8;
  const float sc  = invt[tc + ncol];   // invn applied once per output column
  float lmax = -INFINITY;
#pragma unroll
  for (int r = 0; r < 8; ++r) {
    int eloc = er + mhalf + r;
    float v = acc[r] * sc + mbias[eloc];
    acc[r] = v;
    m[((size_t)b * E + (e0 + eloc)) * T + (t0 + tc + ncol)] = v;
    lmax = fmaxf(lmax, v);
  }
  // block max
  red[tid] = lmax; __syncthreads();
  for (int s = 128; s > 0; s >>= 1) {
    if (tid < s) red[tid] = fmaxf(red[tid], red[tid + s]);
    __syncthreads();
  }
  const float bm = red[0];
  __syncthreads();
  // block sum of exp(v - blockmax) + pass-0 histogram (top byte of encoding)
  float es = 0.f;
#pragma unroll
  for (int r = 0; r < 8; ++r) {
    es += expf(acc[r] - bm);
    atomicAdd(&h[ordenc(acc[r]) >> 24], 1u);
  }
  red[tid] = es; __syncthreads();
  for (int s = 128; s > 0; s >>= 1) {
    if (tid < s) red[tid] += red[tid + s];
    __syncthreads();
  }
  if (h[tid]) atomicAdd(&hist[b * 256 + tid], h[tid]);
  if (tid == 0) {
    int rec = (b * (E / 32) + blockIdx.y) * (T / 64) + blockIdx.x;
    blkmax[rec] = bm;
    blksum[rec] = red[0];
    atomicMax(&bmax_enc[b], ordenc(bm));
  }
}

// ---------------------------------------------------------------------------
// 3) sumexp[b] = sum_blocks blksum * exp(blkmax - gmax)   (deterministic)
// ---------------------------------------------------------------------------
__global__ void combine_sumexp(const float* __restrict__ blkmax, const float* __restrict__ blksum,
                               const unsigned* __restrict__ bmax, float* __restrict__ sumexp) {
  int b = blockIdx.x;
  float gmax = orddec(bmax[b]);
  float s = 0.f;
  for (int i = threadIdx.x; i < BLK_PER_BATCH; i += 256)
    s += blksum[b * BLK_PER_BATCH + i] * expf(blkmax[b * BLK_PER_BATCH + i] - gmax);
  __shared__ float red[256];
  red[threadIdx.x] = s; __syncthreads();
  for (int st = 128; st > 0; st >>= 1) {
    if ((int)threadIdx.x < st) red[threadIdx.x] += red[threadIdx.x + st];
    __syncthreads();
  }
  if (threadIdx.x == 0) sumexp[b] = red[0];
}

// ---------------------------------------------------------------------------
// 4) Radix-select passes 1..3 (pass 0 histogram was fused into the GEMM).
// ---------------------------------------------------------------------------
__global__ void hist_kernel(const float* __restrict__ m, const unsigned* __restrict__ prefix,
                            unsigned* __restrict__ hist, int shift, unsigned himask) {
  int b = blockIdx.y;
  __shared__ unsigned h[256];
  h[threadIdx.x] = 0; __syncthreads();
  unsigned pfx = prefix[b] & himask;
  size_t N = (size_t)E * T;
  const float* mb = m + (size_t)b * N;
  for (size_t i = (size_t)blockIdx.x * blockDim.x + threadIdx.x; i < N;
       i += (size_t)gridDim.x * blockDim.x) {
    unsigned enc = ordenc(mb[i]);
    if ((enc & himask) == pfx) atomicAdd(&h[(enc >> shift) & 0xFFu], 1u);
  }
  __syncthreads();
  if (h[threadIdx.x]) atomicAdd(&hist[b * 256 + threadIdx.x], h[threadIdx.x]);
}

__global__ void scan_kernel(const unsigned* __restrict__ hist, unsigned* __restrict__ prefix,
                            unsigned* __restrict__ kth, int shift) {
  int b = threadIdx.x;
  if (b >= B) return;
  unsigned k = kth[b], pfx = prefix[b];
  for (int d = 255; d >= 0; --d) {
    unsigned c = hist[b * 256 + d];
    if (k <= c) { pfx |= ((unsigned)d) << shift; break; }
    k -= c;
  }
  prefix[b] = pfx; kth[b] = k;
}

// ---------------------------------------------------------------------------
// 5) Gather top-K entries (softmax value + position), flag their t-columns.
// ---------------------------------------------------------------------------
__global__ void gather_kernel(const float* __restrict__ m, const unsigned* __restrict__ prefix,
                              const unsigned* __restrict__ bmax, const float* __restrict__ sumexp,
                              unsigned* __restrict__ cnt, int* __restrict__ top_e,
                              int* __restrict__ top_t, float* __restrict__ top_sp,
                              unsigned* __restrict__ colcount) {
  int b = blockIdx.y;
  unsigned tau = prefix[b];
  float mx = orddec(bmax[b]);
  float inv = 1.0f / sumexp[b];
  size_t N = (size_t)E * T;
  const float* mb = m + (size_t)b * N;
  for (size_t i = (size_t)blockIdx.x * blockDim.x + threadIdx.x; i < N;
       i += (size_t)gridDim.x * blockDim.x) {
    float v = mb[i];
    if (ordenc(v) >= tau) {
      unsigned pos = atomicAdd(&cnt[b], 1u);
      if (pos < (unsigned)K) {
        int e = (int)(i / T), t = (int)(i % T);
        top_e[b * K + pos] = e;
        top_t[b * K + pos] = t;
        top_sp[b * K + pos] = expf(v - mx) * inv;
        atomicAdd(&colcount[b * T + t], 1u);
      }
    }
  }
}

// Deterministic slot assignment: slot = rank of flagged t within batch.
__global__ void compact_cols(const unsigned* __restrict__ colcount, int* __restrict__ colslot,
                             unsigned* __restrict__ ncols) {
  int b = threadIdx.x;
  if (b >= B) return;
  int s = 0;
  for (int t = 0; t < T; ++t)
    if (colcount[b * T + t]) colslot[b * T + t] = s++;
  ncols[b] = (unsigned)s;
}

// ---------------------------------------------------------------------------
// 6) up only at sparse positions: g = sp * (up_w[e,:].x[b,:,t] + up_b[e]);
//    summary[b,e] += g. One wave per entry.
// ---------------------------------------------------------------------------
__global__ void sparse_vals(const float* __restrict__ x, const float* __restrict__ up_w,
                            const float* __restrict__ up_b, const int* __restrict__ top_e,
                            const int* __restrict__ top_t, const float* __restrict__ top_sp,
                            float* __restrict__ top_g, float* __restrict__ summary) {
  int idx = blockIdx.x * 8 + (threadIdx.x >> 5);
  int lane = threadIdx.x & 31;
  if (idx >= B * K) return;
  int b = idx / K;
  int e = top_e[idx], t = top_t[idx];
  float sp = top_sp[idx];
  const float* xp = x + (size_t)b * C * T + t;
  const float* wp = up_w + (size_t)e * C;
  float s = 0.f;
  for (int c = lane; c < C; c += 32) s += wp[c] * xp[(size_t)c * T];
  for (int o = 16; o > 0; o >>= 1) s += __shfl_down(s, o, 32);
  if (lane == 0) {
    float g = sp * (s + up_b[e]);
    top_g[idx] = g;
    atomicAdd(&summary[b * E + e], g);
  }
}

// 7) Sparse down columns: D[b,slot,c] += down_w[c,e]*g. One block per entry.
__global__ void daccum(const int* __restrict__ top_e, const int* __restrict__ top_t,
                       const float* __restrict__ top_g, const float* __restrict__ down_w,
                       const int* __restrict__ colslot, float* __restrict__ D) {
  int idx = blockIdx.x, c = threadIdx.x;
  int b = idx / K;
  int e = top_e[idx], t = top_t[idx];
  float g = top_g[idx];
  int slot = colslot[b * T + t];
  if (slot < 0) return;  // unfilled placeholder entry
  atomicAdd(&D[(((size_t)b * MAXCOLS) + slot) * C + c], down_w[(size_t)c * E + e] * g);
}

// 8) proj[b,c] = summary[b,:].sum_w[c,:] + sum_b[c] + down_b[c]  (down_b folded in)
__global__ void proj_kernel(const float* __restrict__ summary, const float* __restrict__ sum_w,
                            const float* __restrict__ sum_b, const float* __restrict__ down_b,
                            float* __restrict__ proj) {
  int idx = blockIdx.x * 8 + (threadIdx.x >> 5);
  int lane = threadIdx.x & 31;
  if (idx >= B * C) return;
  int b = idx / C, c = idx % C;
  const float* sw = sum_w + (size_t)c * E;
  const float* sm = summary + (size_t)b * E;
  float s = 0.f;
  for (int e = lane; e < E; e += 32) s += sw[e] * sm[e];
  for (int o = 16; o > 0; o >>= 1) s += __shfl_down(s, o, 32);
  if (lane == 0) proj[idx] = s + sum_b[c] + down_b[c];
}

// 9) Sparse corrections to BN stats: for each nonzero column, channel c gets
//    mean_ex[c] += d ; sq_ex[c] += d^2 + 2*proj*d  (since (p+d)^2 - p^2).
__global__ void stats_extra(const float* __restrict__ D, const unsigned* __restrict__ ncols,
                            const float* __restrict__ proj, float* __restrict__ mean_ex,
                            float* __restrict__ sq_ex) {
  int b = blockIdx.x / MAXCOLS, slot = blockIdx.x % MAXCOLS, c = threadIdx.x;
  if (slot >= (int)ncols[b]) return;
  float d = D[(((size_t)b * MAXCOLS) + slot) * C + c];
  atomicAdd(&mean_ex[c], d);
  atomicAdd(&sq_ex[c], d * d + 2.0f * proj[b * C + c] * d);
}

// 10) BN finalize: channel mean/var without ever materializing y.
__global__ void bn_finalize(const float* __restrict__ proj, const float* __restrict__ mean_ex,
                            const float* __restrict__ sq_ex, const float* __restrict__ bn_w,
                            const float* __restrict__ bn_b, float* __restrict__ scale,
                            float* __restrict__ shift) {
  int c = threadIdx.x;
  if (c >= C) return;
  float s = 0.f, sq = 0.f;
  for (int b = 0; b < B; ++b) { float p = proj[b * C + c]; s += p; sq += p * p; }
  float invBT = 1.0f / ((float)B * (float)T);
  float mean = (s * (float)T + mean_ex[c]) * invBT;
  float msq  = (sq * (float)T + sq_ex[c]) * invBT;
  float var  = msq - mean * mean;  // biased variance (matches jnp.var)
  float sc = bn_w[c] * rsqrtf(var + 1e-5f);
  scale[c] = sc;
  shift[c] = bn_b[c] - mean * sc;
}

// 11) out[b,c,t] = (proj[b,c] + sparse) * scale[c] + shift[c]
__global__ void output_kernel(const float* __restrict__ proj, const float* __restrict__ D,
                              const int* __restrict__ colslot, const float* __restrict__ scale,
                              const float* __restrict__ shift, float* __restrict__ out) {
  size_t i = (size_t)blockIdx.x * 256 + threadIdx.x;
  if (i >= (size_t)B * C * T) return;
  int t = (int)(i % T);
  size_t bc = i / T;
  int c = (int)(bc % C), b = (int)(bc / C);
  float v = proj[b * C + c];
  int slot = colslot[b * T + t];
  if (slot >= 0) v += D[(((size_t)b * MAXCOLS) + slot) * C + c];
  out[i] = v * scale[c] + shift[c];
}

// ---------------------------------------------------------------------------
// Host-side orchestration. Workspace needed: ~139 MB (m = 128 MB of it).
// ---------------------------------------------------------------------------
extern "C" void kernel_launch(void* const* d_in, const int* in_sizes, int n_in,
                              void* d_out, int out_size, void* d_ws, size_t ws_size,
                              hipStream_t stream) {
  const float* x      = (const float*)d_in[0];
  const float* up_w   = (const float*)d_in[1];
  const float* up_b   = (const float*)d_in[2];
  const float* mask_w = (const float*)d_in[3];
  const float* mask_b = (const float*)d_in[4];
  const float* sum_w  = (const float*)d_in[5];
  const float* sum_b  = (const float*)d_in[6];
  const float* down_w = (const float*)d_in[7];
  const float* down_b = (const float*)d_in[8];
  const float* bn_w   = (const float*)d_in[9];
  const float* bn_b   = (const float*)d_in[10];

  char* w = (char*)d_ws;
  auto take = [&](size_t bytes) -> char* {
    char* p = w; w += (bytes + 255) & ~(size_t)255; return p;
  };
  float*    m        = (float*)take(sizeof(float) * (size_t)B * E * T);   // 128 MB
  float*    invn     = (float*)take(sizeof(float) * B * T);
  // ---- everything below here is zero-filled each launch ----
  char*     zbase    = w;
  unsigned* bmax     = (unsigned*)take(sizeof(unsigned) * B);
  float*    sumexp   = (float*)take(sizeof(float) * B);
  unsigned* prefix   = (unsigned*)take(sizeof(unsigned) * B);
  unsigned* kth      = (unsigned*)take(sizeof(unsigned) * B);
  unsigned* hist     = (unsigned*)take(sizeof(unsigned) * B * 256);
  unsigned* cnt      = (unsigned*)take(sizeof(unsigned) * B);
  int*      top_e    = (int*)take(sizeof(int) * B * K);
  int*      top_t    = (int*)take(sizeof(int) * B * K);
  float*    top_sp   = (float*)take(sizeof(float) * B * K);
  float*    top_g    = (float*)take(sizeof(float) * B * K);
  float*    summary  = (float*)take(sizeof(float) * B * E);
  float*    proj     = (float*)take(sizeof(float) * B * C);
  unsigned* colcount = (unsigned*)take(sizeof(unsigned) * B * T);
  int*      colslot  = (int*)take(sizeof(int) * B * T);
  unsigned* ncols    = (unsigned*)take(sizeof(unsigned) * B);
  float*    D        = (float*)take(sizeof(float) * (size_t)B * MAXCOLS * C); // 4 MB
  float*    mean_ex  = (float*)take(sizeof(float) * C);
  float*    sq_ex    = (float*)take(sizeof(float) * C);
  float*    scale    = (float*)take(sizeof(float) * C);
  float*    shift    = (float*)take(sizeof(float) * C);
  float*    blkmax   = (float*)take(sizeof(float) * B * BLK_PER_BATCH);   // 64 KB
  float*    blksum   = (float*)take(sizeof(float) * B * BLK_PER_BATCH);   // 64 KB
  char*     zend     = w;

  // Deterministic re-init of all accumulators/counters each call.
  size_t zwords = (size_t)(zend - zbase) / 4;
  fill_u32<<<4096, 256, 0, stream>>>((unsigned*)zbase, zwords, 0u);
  fill_u32<<<1, 256, 0, stream>>>(kth, B, (unsigned)K);
  fill_u32<<<256, 256, 0, stream>>>((unsigned*)colslot, (size_t)B * T, 0xFFFFFFFFu);

  // 1) column norms
  norms_kernel<<<(B * T + 255) / 256, 256, 0, stream>>>(x, invn);

  // 2) WMMA mask GEMM + fused max / blockwise-expsum / pass-0 histogram
  dim3 ggemm(T / 64, E / 32, B);
  mask_gemm_kernel<<<ggemm, 256, 0, stream>>>(x, mask_w, mask_b, invn, m, bmax,
                                              hist, blkmax, blksum);

  // 3) pass-0 scan + softmax denominator from block records
  scan_kernel<<<1, 32, 0, stream>>>(hist, prefix, kth, 24);
  combine_sumexp<<<B, 256, 0, stream>>>(blkmax, blksum, bmax, sumexp);

  // 4) radix-select passes 1..3 (mostly L2-resident re-reads of m)
  const int      shifts[3]  = {16, 8, 0};
  const unsigned himasks[3] = {0xFF000000u, 0xFFFF0000u, 0xFFFFFF00u};
  dim3 gred(1024, B);
  for (int p = 0; p < 3; ++p) {
    fill_u32<<<8, 256, 0, stream>>>(hist, (size_t)B * 256, 0u);
    hist_kernel<<<gred, 256, 0, stream>>>(m, prefix, hist, shifts[p], himasks[p]);
    scan_kernel<<<1, 32, 0, stream>>>(hist, prefix, kth, shifts[p]);
  }

  // 5) gather top-K entries and flag their t-columns
  gather_kernel<<<gred, 256, 0, stream>>>(m, prefix, bmax, sumexp, cnt,
                                          top_e, top_t, top_sp, colcount);
  compact_cols<<<1, 32, 0, stream>>>(colcount, colslot, ncols);

  // 6) sparse up/gated values + summary
  sparse_vals<<<(B * K + 7) / 8, 256, 0, stream>>>(x, up_w, up_b, top_e, top_t,
                                                   top_sp, top_g, summary);
  // 7) sparse down columns
  daccum<<<B * K, 256, 0, stream>>>(top_e, top_t, top_g, down_w, colslot, D);
  // 8) proj (with sum_b and down_b folded in)
  proj_kernel<<<(B * C + 7) / 8, 256, 0, stream>>>(summary, sum_w, sum_b, down_b, proj);
  // 9) sparse BN stat corrections
  stats_extra<<<B * MAXCOLS, 256, 0, stream>>>(D, ncols, proj, mean_ex, sq_ex);
  // 10) BN scale/shift
  bn_finalize<<<1, 256, 0, stream>>>(proj, mean_ex, sq_ex, bn_w, bn_b, scale, shift);
  // 11) final 64 MB write
  output_kernel<<<(int)(((size_t)B * C * T) / 256), 256, 0, stream>>>(
      proj, D, colslot, scale, shift, (float*)d_out);
}